// Averager_87978110091467
// MI455X (gfx1250) — compile-verified
//
#include <hip/hip_runtime.h>
#include <cstdint>

// Problem geometry (fixed by the reference: x is (16, 64, 256, 256) fp32).
constexpr int   W     = 256;
constexpr int   H     = 256;
constexpr float F_LOW = -2.5f;
constexpr float F_UP  =  2.5f;

constexpr int TROWS = 32;          // output rows per block
constexpr int HROWS = TROWS + 2;   // +2 halo rows = 34
// LDS tile: 34 * 256 * 4B = 34,816 bytes (no halo columns; edges are masked)

typedef int v4i __attribute__((vector_size(16)));
typedef __attribute__((address_space(1))) v4i* gas4_t;
typedef __attribute__((address_space(3))) v4i* las4_t;
typedef __attribute__((address_space(1))) int* gas_t;
typedef __attribute__((address_space(3))) int* las_t;

// ---- CDNA5 async global->LDS copies (ASYNCcnt path) -----------------------
__device__ __forceinline__ void async_copy_b128(const float* gsrc, float* ldst) {
#if defined(__gfx1250__)
#if __has_builtin(__builtin_amdgcn_global_load_async_to_lds_b128)
  __builtin_amdgcn_global_load_async_to_lds_b128(
      (gas4_t)(uintptr_t)gsrc, (las4_t)(uintptr_t)ldst, /*offset=*/0, /*cpol=*/0);
#elif __has_builtin(__builtin_amdgcn_global_load_async_to_lds_b32)
  __builtin_amdgcn_global_load_async_to_lds_b32(
      (gas_t)(uintptr_t)(gsrc + 0), (las_t)(uintptr_t)(ldst + 0), 0, 0);
  __builtin_amdgcn_global_load_async_to_lds_b32(
      (gas_t)(uintptr_t)(gsrc + 1), (las_t)(uintptr_t)(ldst + 1), 0, 0);
  __builtin_amdgcn_global_load_async_to_lds_b32(
      (gas_t)(uintptr_t)(gsrc + 2), (las_t)(uintptr_t)(ldst + 2), 0, 0);
  __builtin_amdgcn_global_load_async_to_lds_b32(
      (gas_t)(uintptr_t)(gsrc + 3), (las_t)(uintptr_t)(ldst + 3), 0, 0);
#else
  asm volatile("global_load_async_to_lds_b128 %0, %1, off"
               :
               : "v"((uint32_t)(uintptr_t)ldst), "v"((uint64_t)(uintptr_t)gsrc)
               : "memory");
#endif
#else
  for (int i = 0; i < 4; ++i) ldst[i] = gsrc[i];  // host / non-gfx1250 pass
#endif
}

__device__ __forceinline__ void wait_async_all() {
#if defined(__gfx1250__)
#if __has_builtin(__builtin_amdgcn_s_wait_asynccnt)
  __builtin_amdgcn_s_wait_asynccnt(0);
#else
  asm volatile("s_wait_asynccnt 0" ::: "memory");
#endif
#endif
}

// ---------------------------------------------------------------------------
__global__ __launch_bounds__(256) void averager3x3_kernel(
    const float* __restrict__ x, float* __restrict__ out) {
  __shared__ float tile[HROWS * W];

  const int tid   = threadIdx.x;
  const int bpp   = H / TROWS;                 // row-blocks per plane (8)
  const int plane = blockIdx.x / bpp;
  const int rb    = blockIdx.x - plane * bpp;
  const int row0  = rb * TROWS;                // first output row of this block
  const size_t planeOff = (size_t)plane * (size_t)(H * W);
  const float* src = x + planeOff;

  // Stage halo tile (rows row0-1 .. row0+TROWS, clamped) into LDS via async
  // DMA, 128b per lane: 64 lanes cover one 256-float row, 256 threads stage
  // 4 rows per iteration -> 9 async b128 instructions total.
  {
    const int rowLane = tid >> 6;              // 0..3: which of 4 rows
    const int col4    = (tid & 63) << 2;       // 16B-aligned column group
    for (int tr0 = 0; tr0 < HROWS; tr0 += 4) {
      const int tr = tr0 + rowLane;
      if (tr < HROWS) {
        int gr = row0 + tr - 1;
        gr = gr < 0 ? 0 : (gr > H - 1 ? H - 1 : gr);
        async_copy_b128(src + (size_t)gr * W + col4, &tile[tr * W + col4]);
      }
    }
  }
  wait_async_all();
  __syncthreads();

  // Separable windowed sums: each thread owns one output column. Edge columns
  // read a clamped (duplicate) neighbor whose contribution is masked to zero.
  const int   c   = tid;                               // 0..255
  const int   cm1 = (c > 0)     ? c - 1 : 0;
  const int   cp1 = (c < W - 1) ? c + 1 : W - 1;
  const float okL = (c > 0)     ? 1.0f : 0.0f;
  const float okR = (c < W - 1) ? 1.0f : 0.0f;

  // Horizontal 3-tap masked (sum, count) for tile row tr (global row row0+tr-1).
  auto hsum = [&](int tr, float& s, float& n) {
    const float* rowp = &tile[tr * W];
    float xl = rowp[cm1];
    float xm = rowp[c];
    float xr = rowp[cp1];
    float vl = ((xl > F_LOW) && (xl < F_UP)) ? okL : 0.0f;
    float vm = ((xm > F_LOW) && (xm < F_UP)) ? 1.0f : 0.0f;
    float vr = ((xr > F_LOW) && (xr < F_UP)) ? okR : 0.0f;
    const int grow = row0 + tr - 1;
    const float rowIn = (grow >= 0 && grow < H) ? 1.0f : 0.0f;
    s = (vl * xl + vm * xm + vr * xr) * rowIn;
    n = (vl + vm + vr) * rowIn;
  };

  float s0, n0, s1, n1, s2, n2;
  hsum(0, s0, n0);
  hsum(1, s1, n1);

  float* dst = out + planeOff;
  for (int rr = 0; rr < TROWS; ++rr) {
    hsum(rr + 2, s2, n2);
    float vs   = s0 + s1 + s2;
    float vn   = n0 + n1 + n2;
    float mean = vs / fmaxf(vn, 1.0f);
    float xv   = tile[(rr + 1) * W + c];           // center value
    bool faulty = (xv < F_LOW) || (xv > F_UP);     // strict, NaN passes through
    float result = faulty ? mean : xv;
    // Output is write-once: stream it past the caches (TH=NT).
    __builtin_nontemporal_store(result, &dst[(size_t)(row0 + rr) * W + c]);
    s0 = s1; n0 = n1;
    s1 = s2; n1 = n2;
  }
}

// ---------------------------------------------------------------------------
extern "C" void kernel_launch(void* const* d_in, const int* in_sizes, int n_in,
                              void* d_out, int out_size, void* d_ws, size_t ws_size,
                              hipStream_t stream) {
  (void)n_in; (void)out_size; (void)d_ws; (void)ws_size;
  const float* x  = (const float*)d_in[0];
  float* out      = (float*)d_out;
  const int total = in_sizes[0];                 // 16*64*256*256
  const int planes = total / (H * W);            // 1024
  const int blocks = planes * (H / TROWS);       // 8192
  averager3x3_kernel<<<dim3(blocks), dim3(256), 0, stream>>>(x, out);
}